// MultiModal_Model_decom_text_ts_wf_62234076119117
// MI455X (gfx1250) — compile-verified
//
#include <hip/hip_runtime.h>

// CDNA5 / gfx1250: wave32, WMMA f32 16x16x4 (full-precision f32 matrix path).
typedef float v2f __attribute__((ext_vector_type(2)));
typedef float v8f __attribute__((ext_vector_type(8)));

#define T_DIM   512
#define B_DIM   32
#define TS_FD   76
#define N_WF    200
#define WF_LEN  1000
#define TEXT_D  200
#define WF_D    100
#define HID     256
#define FC_H    256
#define NCLS    2
#define ROWS    (T_DIM * B_DIM)   // 16384

__device__ __forceinline__ v8f wmma_f32_16x16x4(v2f a, v2f b, v8f c) {
    // 8 args: (neg_a, A, neg_b, B, c_mod, C, reuse_a, reuse_b)
    return __builtin_amdgcn_wmma_f32_16x16x4_f32(
        false, a, false, b, (short)0, c, false, false);
}

// Generic tiled GEMM: C[m,n] = act( scale * sum_k A[m,k]*B[k,n] + bias[n] )
// A: M x K (row-major, lda), B: K x N (row-major, ldb), C: M x N (ldc).
// One wave (32 lanes) per 16x16 C tile; blockIdx.z = batch (pointer strides).
// M % 16 == 0 and K % 4 == 0 are required (true for all uses here);
// N edge handled with clamped loads + masked writeback so EXEC is all-1s
// for every v_wmma (ISA 7.12 requirement).
template <int ACT>  // 0 = identity, 1 = tanh
__global__ void gemm_f32_wmma(const float* __restrict__ A,
                              const float* __restrict__ Bm,
                              const float* __restrict__ bias,
                              float* __restrict__ C,
                              int M, int N, int K,
                              int lda, int ldb, int ldc,
                              long strideA, long strideB, long strideC,
                              float scale) {
    A  += (size_t)blockIdx.z * strideA;
    Bm += (size_t)blockIdx.z * strideB;
    C  += (size_t)blockIdx.z * strideC;

    const int m0 = blockIdx.x * 16;
    const int n0 = blockIdx.y * 16;
    const int l  = threadIdx.x;          // 0..31
    const int mrow = l & 15;             // A: lane -> M
    const int kk   = (l >> 4) * 2;       // A/B: half-wave -> K pair (0 or 2)
    const int ncol = l & 15;             // B/C: lane -> N
    const bool nok = (n0 + ncol) < N;
    const int nc   = nok ? (n0 + ncol) : (N - 1);   // clamped, always in-bounds
    const float nmask = nok ? 1.0f : 0.0f;

    v8f acc = {};
    const float* ap = A + (size_t)(m0 + mrow) * lda + kk;
    for (int k0 = 0; k0 < K; k0 += 4) {
        v2f a;
        a.x = ap[k0];
        a.y = ap[k0 + 1];
        const float* bp = Bm + (size_t)(k0 + kk) * ldb + nc;
        v2f b;
        b.x = bp[0]   * nmask;
        b.y = bp[ldb] * nmask;
        acc = wmma_f32_16x16x4(a, b, acc);
    }

    if (!nok) return;  // after all WMMAs; safe to diverge now
    const float bv = bias ? bias[n0 + ncol] : 0.0f;
    const int mb = (l >> 4) * 8;  // C/D: upper half-wave holds M+8
    #pragma unroll
    for (int v = 0; v < 8; ++v) {
        float x = acc[v] * scale + bv;
        if (ACT == 1) x = tanhf(x);
        C[(size_t)(m0 + mb + v) * ldc + (n0 + ncol)] = x;
    }
}

// Fused head: feat = [texts | ts_out | spread] (three contiguous row-major
// segments, ld = 200/256/100, same linear row index 0..16383), then
// h = relu(feat @ W1 + b1) kept entirely in registers/LDS, then
// out = h @ W2 + b2 (N=2). Block = 4 waves; wave w owns h columns 64w..64w+63
// (4 WMMA N-tiles). h (16 MB) never goes to HBM.
__global__ void head_fused_wmma(const float* __restrict__ texts,   // ROWS x 200
                                const float* __restrict__ ts_out,  // ROWS x 256
                                const float* __restrict__ spread,  // ROWS x 100
                                const float* __restrict__ W1,      // 556 x 256
                                const float* __restrict__ b1,      // 256
                                const float* __restrict__ W2,      // 256 x 2
                                const float* __restrict__ b2,      // 2
                                float* __restrict__ out) {         // ROWS x 2
    __shared__ float h_lds[16][FC_H + 1];   // +1 pad: avoid bank conflicts

    const int m0   = blockIdx.x * 16;
    const int wave = threadIdx.x >> 5;      // 0..3
    const int l    = threadIdx.x & 31;
    const int mrow = l & 15;
    const int kk   = (l >> 4) * 2;
    const int ncol = l & 15;

    v8f acc[4];
    #pragma unroll
    for (int t = 0; t < 4; ++t) acc[t] = (v8f){};

    const float* segA[3] = { texts  + (size_t)m0 * TEXT_D,
                             ts_out + (size_t)m0 * HID,
                             spread + (size_t)m0 * WF_D };
    const int segLd[3] = { TEXT_D, HID, WF_D };   // each % 4 == 0

    int kbase = 0;
    for (int s = 0; s < 3; ++s) {
        const float* Ab = segA[s] + (size_t)mrow * segLd[s] + kk;
        const int Ks = segLd[s];
        for (int k0 = 0; k0 < Ks; k0 += 4) {
            v2f a;
            a.x = Ab[k0];
            a.y = Ab[k0 + 1];
            __builtin_prefetch(Ab + k0 + 32, 0, 3);   // global_prefetch_b8
            const float* Bp = W1 + (size_t)(kbase + k0 + kk) * FC_H + ncol;
            #pragma unroll
            for (int t = 0; t < 4; ++t) {
                const int n0 = (wave * 4 + t) * 16;
                v2f b;
                b.x = Bp[n0];
                b.y = Bp[FC_H + n0];
                acc[t] = wmma_f32_16x16x4(a, b, acc[t]);
            }
        }
        kbase += segLd[s];
    }

    // bias + ReLU -> LDS h tile (each wave writes its own 64 columns)
    const int mb = (l >> 4) * 8;
    #pragma unroll
    for (int t = 0; t < 4; ++t) {
        const int n0 = (wave * 4 + t) * 16;
        const float bv = b1[n0 + ncol];
        #pragma unroll
        for (int v = 0; v < 8; ++v)
            h_lds[mb + v][n0 + ncol] = fmaxf(acc[t][v] + bv, 0.0f);
    }
    __syncthreads();

    // out = h @ W2 + b2 : 16 rows x 2 classes, one lane per (row, cls)
    if (threadIdx.x < 32) {
        const int r = threadIdx.x & 15;
        const int c = threadIdx.x >> 4;
        float s = b2[c];
        for (int k = 0; k < FC_H; ++k)
            s += h_lds[r][k] * W2[k * NCLS + c];
        out[(size_t)(m0 + r) * NCLS + c] = s;
    }
}

extern "C" void kernel_launch(void* const* d_in, const int* in_sizes, int n_in,
                              void* d_out, int out_size, void* d_ws, size_t ws_size,
                              hipStream_t stream) {
    (void)in_sizes; (void)n_in; (void)out_size; (void)ws_size;
    const float* texts    = (const float*)d_in[0];   // (512,32,200)
    const float* ts       = (const float*)d_in[1];   // (512,32,76)
    const float* waveform = (const float*)d_in[2];   // (32,200,1000)
    const float* wmat     = (const float*)d_in[3];   // (32,512,200)
    const float* W_wf     = (const float*)d_in[4];   // (1000,100)
    const float* b_wf     = (const float*)d_in[5];
    const float* W_ts     = (const float*)d_in[6];   // (76,256)
    const float* b_ts     = (const float*)d_in[7];
    const float* W1       = (const float*)d_in[8];   // (556,256)
    const float* b1       = (const float*)d_in[9];
    const float* W2       = (const float*)d_in[10];  // (256,2)
    const float* b2       = (const float*)d_in[11];
    float* out = (float*)d_out;

    // workspace layout (f32): wf_out | ts_out | spread  (~25.9 MB)
    float* wf_out = (float*)d_ws;                                  // 6400*100
    float* ts_out = wf_out + (size_t)(B_DIM * N_WF) * WF_D;        // 16384*256
    float* spread = ts_out + (size_t)ROWS * HID;                   // 16384*100

    // 1) wf_out = waveform @ W_wf + b_wf   : (6400,1000)x(1000,100)
    gemm_f32_wmma<0><<<dim3((B_DIM * N_WF) / 16, (WF_D + 15) / 16, 1), 32, 0, stream>>>(
        waveform, W_wf, b_wf, wf_out,
        B_DIM * N_WF, WF_D, WF_LEN, WF_LEN, WF_D, WF_D, 0, 0, 0, 1.0f);

    // 2) ts_out = tanh(ts @ W_ts + b_ts)   : (16384,76)x(76,256)
    gemm_f32_wmma<1><<<dim3(ROWS / 16, HID / 16, 1), 32, 0, stream>>>(
        ts, W_ts, b_ts, ts_out,
        ROWS, HID, TS_FD, TS_FD, HID, HID, 0, 0, 0, 1.0f);

    // 3) spread[b] = wmat[b] @ wf_out[b] / 200 : 32 x (512,200)x(200,100)
    gemm_f32_wmma<0><<<dim3(T_DIM / 16, (WF_D + 15) / 16, B_DIM), 32, 0, stream>>>(
        wmat, wf_out, nullptr, spread,
        T_DIM, WF_D, N_WF, N_WF, WF_D, WF_D,
        (long)T_DIM * N_WF, (long)N_WF * WF_D, (long)T_DIM * WF_D,
        1.0f / (float)N_WF);

    // 4+5) fused concat-GEMM + ReLU + tiny head GEMM
    head_fused_wmma<<<ROWS / 16, 128, 0, stream>>>(
        texts, ts_out, spread, W1, b1, W2, b2, out);
}